// GraphEncoder_14388140442144
// MI455X (gfx1250) — compile-verified
//
#include <hip/hip_runtime.h>
#include <hip/hip_bf16.h>

typedef __attribute__((ext_vector_type(2))) float v2f;
typedef __attribute__((ext_vector_type(8))) float v8f;

// ---------------------------------------------------------------------------
// Degree / normalization
// ---------------------------------------------------------------------------
__global__ void k_deg_init(float* __restrict__ deg, int N) {
    int i = blockIdx.x * blockDim.x + threadIdx.x;
    if (i < N) deg[i] = 1.0f;  // self-loop contributes 1 to every node
}

__global__ void k_deg_edges(const long long* __restrict__ ei, int E,
                            float* __restrict__ deg) {
    int e = blockIdx.x * blockDim.x + threadIdx.x;
    if (e < E) {
        int d = (int)ei[(size_t)E + e];  // dst row of edge_index
        atomicAdd(&deg[d], 1.0f);
    }
}

__global__ void k_rsqrt_inplace(float* __restrict__ deg, int N) {
    int i = blockIdx.x * blockDim.x + threadIdx.x;
    if (i < N) deg[i] = rsqrtf(deg[i]);
}

// ---------------------------------------------------------------------------
// Fused GCN GEMM via V_WMMA_F32_16X16X4_F32 (fp32 matrix pipe, wave32).
//
//   H[N,128] = act(A)[N,K] @ W[K,128]           (gather source for scatter)
//   O[N,128] = H * dinv[row]^2 (+ bout)         (self-loop-initialized accum)
//
// ACT_IN:  apply relu(a + bin[k]) to the A operand on load (fuses the
//          previous layer's bias+ReLU epilogue into this GEMM's prologue).
// BIAS_OUT: add bout[col] to O (bias commutes with the scatter-sum).
//
// One block = 256 threads = 8 waves; wave w owns the 16x16 output tile at
// rows [blockIdx.x*16,+16) x cols [16w,+16).
// A fragment (16x4 f32): lanes 0-15 hold M=lane,K={k,k+1}; lanes 16-31 hold
// M=lane-16,K={k+2,k+3}.  C/D (16x16 f32, 8 VGPRs): VGPR r -> row r
// (lanes 0-15) / row r+8 (lanes 16-31), N = lane&15.
// ---------------------------------------------------------------------------
template <int K, bool ACT_IN, bool BIAS_OUT>
__global__ void k_gcn_gemm_wmma(const float* __restrict__ A,
                                const float* __restrict__ W,
                                const float* __restrict__ bin,
                                const float* __restrict__ bout,
                                const float* __restrict__ dinv,
                                float* __restrict__ H,
                                float* __restrict__ O, int N) {
    const int wave   = threadIdx.x >> 5;
    const int lane   = threadIdx.x & 31;
    const int r0     = blockIdx.x * 16;
    const int c0     = wave * 16;
    const int m      = lane & 15;
    const int hiHalf = lane >> 4;        // 0 for lanes 0-15, 1 for 16-31
    const int kh     = hiHalf << 1;      // K offset within 4-wide step
    const int n      = lane & 15;

    const float* __restrict__ arow = A + (size_t)(r0 + m) * K;

    v8f acc = {};
#pragma unroll
    for (int k = 0; k < K; k += 4) {
        const int kb = k + kh;
        float ax = arow[kb];
        float ay = arow[kb + 1];
        if (ACT_IN) {
            ax = fmaxf(ax + bin[kb], 0.0f);
            ay = fmaxf(ay + bin[kb + 1], 0.0f);
        }
        v2f a;
        a.x = ax;
        a.y = ay;
        v2f b;
        b.x = W[(size_t)kb * 128 + c0 + n];
        b.y = W[(size_t)(kb + 1) * 128 + c0 + n];
        acc = __builtin_amdgcn_wmma_f32_16x16x4_f32(
            /*neg_a=*/false, a, /*neg_b=*/false, b,
            /*c_mod=*/(short)0, acc, /*reuse_a=*/false, /*reuse_b=*/false);
    }

    const int rbase = r0 + (hiHalf << 3);   // +0 or +8
    const int col   = c0 + (lane & 15);
    const float bo  = BIAS_OUT ? bout[col] : 0.0f;
#pragma unroll
    for (int r = 0; r < 8; ++r) {
        const int row = rbase + r;
        const float h = acc[r];
        H[(size_t)row * 128 + col] = h;
        const float dv = dinv[row];
        float o = h * dv * dv;
        if (BIAS_OUT) o += bo;
        O[(size_t)row * 128 + col] = o;
    }
}

// ---------------------------------------------------------------------------
// Edge scatter: O[dst] += H[src] * (dinv[src]*dinv[dst])
// One wave (32 lanes) per edge; each lane handles 4 consecutive features.
// H rows are L2-resident (51 MB << 192 MB L2); scatter uses
// global_atomic_add_f32 (no packed f32 atomic exists -> 4 scalar atomics).
// ---------------------------------------------------------------------------
__global__ void k_edge_scatter(const long long* __restrict__ ei, int E,
                               const float* __restrict__ H,
                               const float* __restrict__ dinv,
                               float* __restrict__ O) {
    const int lane = threadIdx.x & 31;
    const int sub  = threadIdx.x >> 5;               // wave index in block (0..7)
    const int e    = blockIdx.x * 8 + sub;
    if (e >= E) return;

    const int s = (int)ei[e];
    const int d = (int)ei[(size_t)E + e];
    const float norm = dinv[s] * dinv[d];

    const float4 h = ((const float4*)(H + (size_t)s * 128))[lane];
    float* op = O + (size_t)d * 128 + lane * 4;
    atomicAdd(op + 0, h.x * norm);
    atomicAdd(op + 1, h.y * norm);
    atomicAdd(op + 2, h.z * norm);
    atomicAdd(op + 3, h.w * norm);
}

// ---------------------------------------------------------------------------
// Launch
// ---------------------------------------------------------------------------
extern "C" void kernel_launch(void* const* d_in, const int* in_sizes, int n_in,
                              void* d_out, int out_size, void* d_ws, size_t ws_size,
                              hipStream_t stream) {
    const float*     x   = (const float*)d_in[0];      // [N,32]
    const long long* ei  = (const long long*)d_in[1];  // [2,E] int64
    const float*     W1  = (const float*)d_in[2];      // [32,128]
    const float*     b1  = (const float*)d_in[3];      // [128]
    const float*     W2  = (const float*)d_in[4];      // [128,128]
    const float*     b2  = (const float*)d_in[5];      // [128]
    float*           out = (float*)d_out;              // [N,128]

    const int N = in_sizes[0] / 32;   // 100000 (divisible by 16)
    const int E = in_sizes[1] / 2;    // 1600000 (divisible by 8)

    char* ws = (char*)d_ws;
    float* dinv = (float*)ws;                                   // N floats
    float* bufA = (float*)(ws + (((size_t)N * 4 + 255) & ~(size_t)255));
    float* bufB = bufA + (size_t)N * 128;                       // N*128 each

    const int TPB = 256;

    // --- normalization coefficients ---------------------------------------
    k_deg_init<<<(N + TPB - 1) / TPB, TPB, 0, stream>>>(dinv, N);
    k_deg_edges<<<(E + TPB - 1) / TPB, TPB, 0, stream>>>(ei, E, dinv);
    k_rsqrt_inplace<<<(N + TPB - 1) / TPB, TPB, 0, stream>>>(dinv, N);

    // --- layer 1: H1 = x@W1; bufB = H1*dinv^2 (self-loop init) ------------
    k_gcn_gemm_wmma<32, false, false><<<N / 16, TPB, 0, stream>>>(
        x, W1, nullptr, nullptr, dinv, bufA, bufB, N);
    //     bufB += norm * H1[src]   (aggregation; b1/ReLU deferred)
    k_edge_scatter<<<(E + 7) / 8, TPB, 0, stream>>>(ei, E, bufA, dinv, bufB);

    // --- layer 2: A = relu(bufB + b1) on the fly; H2 = A@W2;
    //              out = H2*dinv^2 + b2 (self-loop init + bias) -------------
    k_gcn_gemm_wmma<128, true, true><<<N / 16, TPB, 0, stream>>>(
        bufB, W2, b1, b2, dinv, bufA, out, N);
    //     out += norm * H2[src]
    k_edge_scatter<<<(E + 7) / 8, TPB, 0, stream>>>(ei, E, bufA, dinv, out);
}